// TemporalAttention_53446573031792
// MI455X (gfx1250) — compile-verified
//
#include <hip/hip_runtime.h>
#include <hip/hip_bf16.h>

typedef __attribute__((ext_vector_type(2))) float v2f;
typedef __attribute__((ext_vector_type(8))) float v8f;

#define TT 24      // T
#define NN 325     // N
#define DD 64      // D
#define PROW 32    // padded row count (2 WMMA m-tiles)
#define LDW 68     // LDS row stride (floats), padded to dodge bank conflicts

// One 16x16 output tile of  D(16x16) = A(16x64 from LDS) * B(64x16 from global)
// using V_WMMA_F32_16X16X4_F32 over 16 k-steps of 4.
// A layout (16x4 f32): lanes 0-15 -> rows M=0..15 with VGPR0=K0,VGPR1=K1;
//                      lanes 16-31 -> same rows with K2,K3.
// B layout (4x16 f32): VGPR0 = {K0 (lanes 0-15), K2 (lanes 16-31)}, VGPR1 = {K1,K3}.
// C/D layout: VGPR r -> row (r + 8*(lane>>4)), col = lane&15.
__device__ __forceinline__ v8f tile_gemm16(const float* __restrict__ A_lds,   // [PROW][LDW]
                                           const float* __restrict__ W,       // 64x64 row-major [k][n]
                                           int mtile, int ntile, int lane)
{
    v8f acc = {};
    const int m    = lane & 15;
    const int arow = mtile * 16 + m;            // 0..31, pad rows hold zeros/garbage
    const int kh   = (lane >> 4) << 1;          // 0 or 2
    const int col  = ntile * 16 + m;
#pragma unroll
    for (int ks = 0; ks < 16; ++ks) {
        const int kk = ks * 4 + kh;
        v2f a, b;
        a.x = A_lds[arow * LDW + kk];
        a.y = A_lds[arow * LDW + kk + 1];
        b.x = W[kk * DD + col];
        b.y = W[(kk + 1) * DD + col];
        acc = __builtin_amdgcn_wmma_f32_16x16x4_f32(
                  false, a, false, b, (short)0, acc, false, false);
    }
    return acc;
}

__global__ __launch_bounds__(256) void temporal_attn_fused(
    const float* __restrict__ X,
    const float* __restrict__ Wq, const float* __restrict__ bq,
    const float* __restrict__ Wk, const float* __restrict__ bk,
    const float* __restrict__ Wv, const float* __restrict__ bv,
    const float* __restrict__ Wf1, const float* __restrict__ bf1,
    const float* __restrict__ Wf2, const float* __restrict__ bf2,
    float* __restrict__ out)
{
    // Xs doubles as the final-result staging buffer (dead after phase 2,
    // rewritten in phase 5 epilogue; phase barriers order the reuse).
    __shared__ float Xs [PROW * LDW];
    __shared__ float qs [PROW * LDW];
    __shared__ float kss[PROW * LDW];
    __shared__ float vss[PROW * LDW];
    __shared__ float os [PROW * LDW];
    __shared__ float hs [PROW * LDW];

    const int blk  = blockIdx.x;       // b * N + n
    const int b    = blk / NN;
    const int n    = blk - b * NN;
    const int tid  = threadIdx.x;
    const int lane = tid & 31;
    const int wave = tid >> 5;

    // ---- Phase 1: stage X(b, :, n, :) into LDS; zero the 8 pad rows of Xs/os ----
    for (int i = tid; i < PROW * DD; i += 256) {
        const int t = i >> 6, d0 = i & 63;
        float v = 0.0f;
        if (t < TT) v = X[(((size_t)b * TT + t) * NN + n) * DD + d0];
        Xs[t * LDW + d0] = v;
        if (t >= TT) os[t * LDW + d0] = 0.0f;   // pad rows read by FFN1 A-fragments
    }
    __syncthreads();

    // ---- Phase 2: q,k,v = relu(X@W + b).  24 WMMA tiles, 3 per wave ----
#pragma unroll
    for (int it = 0; it < 3; ++it) {
        const int task  = wave + it * 8;        // 0..23, uniform per wave
        const int mat   = task >> 3;            // 0=q 1=k 2=v
        const int tile  = task & 7;
        const int mtile = tile >> 2, ntile = tile & 3;
        const float* W    = (mat == 0) ? Wq : (mat == 1) ? Wk : Wv;
        const float* bias = (mat == 0) ? bq : (mat == 1) ? bk : bv;
        float*       dst  = (mat == 0) ? qs : (mat == 1) ? kss : vss;
        v8f acc = tile_gemm16(Xs, W, mtile, ntile, lane);
        const int col = ntile * 16 + (lane & 15);
        const float bc = bias[col];
#pragma unroll
        for (int r = 0; r < 8; ++r) {           // unconditional: rows 24..31 are scratch
            const int row = mtile * 16 + r + ((lane >> 4) << 3);
            const float v = acc[r] + bc;
            dst[row * LDW + col] = v > 0.f ? v : 0.f;
        }
    }
    __syncthreads();

    // ---- Phase 3: per-head triu-masked softmax attention; wave=head, lane=row t ----
    {
        const int c0 = wave * 8;                // head feature offset
        if (lane < TT) {
            const int t = lane;
            float qrow[8];
#pragma unroll
            for (int j = 0; j < 8; ++j) qrow[j] = qs[t * LDW + c0 + j];
            float S[TT];
            float mx = -3.0e38f;
#pragma unroll
            for (int s = 0; s < TT; ++s) {
                float sc;
                if (s >= t) {                   // triu kept
                    sc = 0.f;
#pragma unroll
                    for (int j = 0; j < 8; ++j) sc += qrow[j] * kss[s * LDW + c0 + j];
                    sc *= 0.35355339059327373f; // 1/sqrt(8)
                } else {
                    sc = -32767.0f;             // -2^15 + 1
                }
                S[s] = sc;
                mx = fmaxf(mx, sc);
            }
            float sum = 0.f;
#pragma unroll
            for (int s = 0; s < TT; ++s) { float e = __expf(S[s] - mx); S[s] = e; sum += e; }
            const float inv = 1.0f / sum;
            float o[8] = {};
#pragma unroll
            for (int s = 0; s < TT; ++s) {
                const float w = S[s] * inv;
#pragma unroll
                for (int j = 0; j < 8; ++j) o[j] += w * vss[s * LDW + c0 + j];
            }
#pragma unroll
            for (int j = 0; j < 8; ++j) os[t * LDW + c0 + j] = o[j];
        }
    }
    __syncthreads();

    // ---- Phase 4: h = relu(out @ Wf1 + bf1).  8 tiles, 1 per wave ----
    {
        const int mtile = wave >> 2, ntile = wave & 3;
        v8f acc = tile_gemm16(os, Wf1, mtile, ntile, lane);
        const int col = ntile * 16 + (lane & 15);
        const float bc = bf1[col];
#pragma unroll
        for (int r = 0; r < 8; ++r) {
            const int row = mtile * 16 + r + ((lane >> 4) << 3);
            const float v = acc[r] + bc;
            hs[row * LDW + col] = v > 0.f ? v : 0.f;
        }
    }
    __syncthreads();

    // ---- Phase 5: result = h @ Wf2 + bf2 -> stage into Xs ----
    {
        const int mtile = wave >> 2, ntile = wave & 3;
        v8f acc = tile_gemm16(hs, Wf2, mtile, ntile, lane);
        const int col = ntile * 16 + (lane & 15);
        const float bc = bf2[col];
#pragma unroll
        for (int r = 0; r < 8; ++r) {
            const int row = mtile * 16 + r + ((lane >> 4) << 3);
            Xs[row * LDW + col] = acc[r] + bc;
        }
    }
    __syncthreads();

    // ---- Phase 6: cooperative, coalesced store of the valid 24x64 region ----
    for (int i = tid; i < TT * DD; i += 256) {
        const int t = i >> 6, d0 = i & 63;
        out[(((size_t)b * TT + t) * NN + n) * DD + d0] = Xs[t * LDW + d0];
    }
}

extern "C" void kernel_launch(void* const* d_in, const int* in_sizes, int n_in,
                              void* d_out, int out_size, void* d_ws, size_t ws_size,
                              hipStream_t stream) {
    (void)in_sizes; (void)n_in; (void)out_size; (void)d_ws; (void)ws_size;
    const float* X   = (const float*)d_in[0];
    // d_in[1] = STE (unused by the reference computation)
    const float* Wq  = (const float*)d_in[2];
    const float* bq  = (const float*)d_in[3];
    const float* Wk  = (const float*)d_in[4];
    const float* bk  = (const float*)d_in[5];
    const float* Wv  = (const float*)d_in[6];
    const float* bv  = (const float*)d_in[7];
    const float* Wf1 = (const float*)d_in[8];
    const float* bf1 = (const float*)d_in[9];
    const float* Wf2 = (const float*)d_in[10];
    const float* bf2 = (const float*)d_in[11];
    float* out = (float*)d_out;

    dim3 grid(32 * NN);   // one block per (b, n)
    dim3 block(256);      // 8 wave32 waves
    temporal_attn_fused<<<grid, block, 0, stream>>>(
        X, Wq, bq, Wk, bk, Wv, bv, Wf1, bf1, Wf2, bf2, out);
}